// Hypernet_trans_48077863911466
// MI455X (gfx1250) — compile-verified
//
#include <hip/hip_runtime.h>
#include <hip/hip_bf16.h>
#include <math.h>

// ---------------------------------------------------------------------------
// Hypernet transformer forward, M=2 GEMV chain, fp32, CDNA5 (gfx1250).
// Heavy layers = bandwidth-bound GEMVs on V_WMMA_F32_16X16X4_F32:
//   A = 16x4 activation tile from a 16-row zero-padded activation buffer,
//   B = 4x16 weight tile, one contiguous non-temporal b64 per lane.
// ---------------------------------------------------------------------------

#define DIM   4096
#define EDIM  8192
#define HEADD 2048   // DIM / 2 heads

typedef __attribute__((ext_vector_type(2))) float v2f;
typedef __attribute__((ext_vector_type(4))) float v4f;
typedef __attribute__((ext_vector_type(8))) float v8f;

// ---------------------------------------------------------------------------
// Zero the padded activation buffers (rows 2..15 must read as 0 for WMMA A).
// ---------------------------------------------------------------------------
__global__ __launch_bounds__(256) void zero_kernel(float* __restrict__ p, int n4)
{
    int i = blockIdx.x * blockDim.x + threadIdx.x;
    if (i < n4) reinterpret_cast<v4f*>(p)[i] = (v4f){0.f, 0.f, 0.f, 0.f};
}

// ---------------------------------------------------------------------------
// Embedding: x0[0][d] = relu(ray0*emb1_w[d] + emb1_b[d]); row 1 likewise.
// Writes rows 0,1 of the padded [16, DIM] activation buffer.
// ---------------------------------------------------------------------------
__global__ __launch_bounds__(256) void embed_kernel(
    const float* __restrict__ ray,
    const float* __restrict__ w1, const float* __restrict__ b1,
    const float* __restrict__ w2, const float* __restrict__ b2,
    float* __restrict__ x0)
{
    int i = blockIdx.x * blockDim.x + threadIdx.x;
    if (i < DIM) {
        float r0 = ray[0], r1 = ray[1];
        x0[i]       = fmaxf(r0 * w1[i] + b1[i], 0.f);
        x0[DIM + i] = fmaxf(r1 * w2[i] + b2[i], 0.f);
    }
}

// ---------------------------------------------------------------------------
// GEMV via WMMA f32 16x16x4, K compile-time (constant trip count -> unroll).
//   out[2,N] = A[16,K](rows 0,1 live) @ W[N,K]^T + bias (+ resid) (+ relu)
// Block = 16 waves (512 thr) = one 16-column tile; wave w owns K/16 slice.
// A-frag (16x4): lane m<16 -> A[m][k..k+1]; lane m+16 -> A[m][k+2..k+3]
//   (plain b64 load from the zero-padded 16-row buffer; EXEC all-ones).
// B-frag (4x16): lane l<16 -> W[n0+l][k..k+1]; lane l+16 -> W[n0+l][k+2..k+3]
//   = one contiguous non-temporal global_load_b64 per lane (read-once stream).
// C/D (16x16): VGPR r, lanes 0-15 = row M=r -> rows 0,1 live in c[0],c[1].
// Two accumulators (k, k+4) break the WMMA accumulator RAW chain.
// ---------------------------------------------------------------------------
template <int K, bool RELU, bool HAS_RESID>
__global__ __launch_bounds__(512) void gemv_wmma_kernel(
    const float* __restrict__ A,       // [16, K] padded, rows 2..15 zero
    const float* __restrict__ W,       // [N, K]
    const float* __restrict__ bias,    // [N]
    const float* __restrict__ resid,   // [2, ldOut] (unused unless HAS_RESID)
    float* __restrict__ out,           // [2, ldOut]
    int N, int ldOut)
{
    __shared__ float red[16][2][16];

    const int tid  = threadIdx.x;
    const int lane = tid & 31;
    const int wv   = tid >> 5;              // 0..15
    const int n0   = blockIdx.x * 16;
    const int lm   = lane & 15;
    const int koff = (lane >> 4) * 2;       // 0 for lanes 0-15, 2 for 16-31

    constexpr int KSLICE = K >> 4;          // 256 for K=4096
    const int k0 = wv * KSLICE;

    const float* arow = A + (size_t)lm * K + koff;          // padded rows
    const float* wrow = W + (size_t)(n0 + lm) * K + koff;

    v8f c0 = {};
    v8f c1 = {};

#pragma unroll 8
    for (int kk = 0; kk < KSLICE; kk += 8) {
        const int k = k0 + kk;
        // weight fragments: streaming, read-once -> non-temporal
        v2f b0 = __builtin_nontemporal_load(
                     reinterpret_cast<const v2f*>(wrow + k));
        v2f b1 = __builtin_nontemporal_load(
                     reinterpret_cast<const v2f*>(wrow + k + 4));
        // activation fragments: tiny, cache-resident, already zero-padded
        v2f a0 = *reinterpret_cast<const v2f*>(arow + k);
        v2f a1 = *reinterpret_cast<const v2f*>(arow + k + 4);

        c0 = __builtin_amdgcn_wmma_f32_16x16x4_f32(
                 false, a0, false, b0, (short)0, c0, false, false);
        c1 = __builtin_amdgcn_wmma_f32_16x16x4_f32(
                 false, a1, false, b1, (short)0, c1, false, false);
    }

    if (lane < 16) {                 // rows M=0 and M=1 live in c[0], c[1]
        red[wv][0][lane] = c0[0] + c1[0];
        red[wv][1][lane] = c0[1] + c1[1];
    }
    __syncthreads();

    if (tid < 32) {
        int row = tid >> 4, col = tid & 15;
        float s = 0.f;
#pragma unroll
        for (int w = 0; w < 16; ++w) s += red[w][row][col];
        int n = n0 + col;
        s += bias[n];
        if (HAS_RESID) s += resid[(size_t)row * ldOut + n];
        if (RELU)      s = fmaxf(s, 0.f);
        out[(size_t)row * ldOut + n] = s;
    }
}

// ---------------------------------------------------------------------------
// Attention core: scores (8 dots of len 2048), softmax over m (2 entries),
// o[l,d] = sum_m p[h(d)][l][m] * v[m][d].  One block, 8 waves.
// Writes rows 0,1 of the padded [16, DIM] output buffer.
// ---------------------------------------------------------------------------
__global__ __launch_bounds__(256) void attn_kernel(
    const float* __restrict__ qkv,   // [2, 3*DIM]
    float* __restrict__ o)           // [16, DIM] padded, writes rows 0,1
{
    __shared__ float sc[8];
    __shared__ float p[2][2][2];     // [h][l][m]

    const int tid = threadIdx.x, lane = tid & 31, wv = tid >> 5;
    const int h = (wv >> 2) & 1, l = (wv >> 1) & 1, m = wv & 1;

    const float* q  = qkv + (size_t)l * 3 * DIM + h * HEADD;
    const float* kk = qkv + (size_t)m * 3 * DIM + DIM + h * HEADD;

    float s = 0.f;
    for (int j = lane; j < HEADD; j += 32) s += q[j] * kk[j];
#pragma unroll
    for (int off = 16; off; off >>= 1) s += __shfl_xor(s, off, 32);
    if (lane == 0) sc[wv] = s * (1.0f / sqrtf((float)HEADD));
    __syncthreads();

    if (tid < 4) {
        int hh = tid >> 1, ll = tid & 1;
        float s0 = sc[(hh << 2) | (ll << 1) | 0];
        float s1 = sc[(hh << 2) | (ll << 1) | 1];
        float mx = fmaxf(s0, s1);
        float e0 = expf(s0 - mx), e1 = expf(s1 - mx);
        float inv = 1.f / (e0 + e1);
        p[hh][ll][0] = e0 * inv;
        p[hh][ll][1] = e1 * inv;
    }
    __syncthreads();

    const float* v0 = qkv + 2 * DIM;
    const float* v1 = qkv + (size_t)3 * DIM + 2 * DIM;
    for (int idx = tid; idx < 2 * DIM; idx += 256) {
        int ll = idx / DIM, d = idx - ll * DIM, hh = d / HEADD;
        o[(size_t)ll * DIM + d] = p[hh][ll][0] * v0[d] + p[hh][ll][1] * v1[d];
    }
}

// ---------------------------------------------------------------------------
// Final expert output layer + mean over seq: out = 0.5*dot(a0+a1, w2) + b2
// ---------------------------------------------------------------------------
__global__ __launch_bounds__(256) void final_kernel(
    const float* __restrict__ a,     // [2, EDIM]
    const float* __restrict__ w2,    // [EDIM]
    const float* __restrict__ b2,    // [1]
    float* __restrict__ out)
{
    __shared__ float red[256];
    int tid = threadIdx.x;
    float s = 0.f;
    for (int j = tid; j < EDIM; j += 256) s += (a[j] + a[EDIM + j]) * w2[j];
    red[tid] = s;
    __syncthreads();
#pragma unroll
    for (int off = 128; off; off >>= 1) {
        if (tid < off) red[tid] += red[tid + off];
        __syncthreads();
    }
    if (tid == 0) out[0] = 0.5f * red[0] + b2[0];
}

// ---------------------------------------------------------------------------
extern "C" void kernel_launch(void* const* d_in, const int* in_sizes, int n_in,
                              void* d_out, int out_size, void* d_ws, size_t ws_size,
                              hipStream_t stream)
{
    (void)in_sizes; (void)n_in; (void)out_size; (void)ws_size;

    const float* ray        = (const float*)d_in[0];
    const float* emb1_w     = (const float*)d_in[1];
    const float* emb1_b     = (const float*)d_in[2];
    const float* emb2_w     = (const float*)d_in[3];
    const float* emb2_b     = (const float*)d_in[4];
    const float* attn_in_w  = (const float*)d_in[5];
    const float* attn_in_b  = (const float*)d_in[6];
    const float* attn_out_w = (const float*)d_in[7];
    const float* attn_out_b = (const float*)d_in[8];
    const float* ffn1_w     = (const float*)d_in[9];
    const float* ffn1_b     = (const float*)d_in[10];
    const float* ffn2_w     = (const float*)d_in[11];
    const float* ffn2_b     = (const float*)d_in[12];
    const float* exp_w1     = (const float*)d_in[13];
    const float* exp_b1     = (const float*)d_in[14];
    const float* exp_w2     = (const float*)d_in[15];
    const float* exp_b2     = (const float*)d_in[16];
    // d_in[17] = idx (single expert, always 0) — unused.

    // Workspace: five 16-row padded activation buffers first (zeroed in one
    // contiguous shot), then the plain [2,*] buffers.
    float* ws   = (float*)d_ws;
    float* x0p  = ws;                   // [16, DIM]
    float* op   = x0p  + 16 * DIM;      // [16, DIM]
    float* x1p  = op   + 16 * DIM;      // [16, DIM]
    float* hp   = x1p  + 16 * DIM;      // [16, DIM]
    float* x2p  = hp   + 16 * DIM;      // [16, DIM]
    float* qkv  = x2p  + 16 * DIM;      // [2, 3*DIM]
    float* abuf = qkv  + 6 * DIM;       // [2, EDIM]

    float* out = (float*)d_out;

    // 0. zero the padded region (rows 2..15 must be 0 for the WMMA A-tiles)
    {
        int n4 = (5 * 16 * DIM) / 4;    // float4 count
        zero_kernel<<<(n4 + 255) / 256, 256, 0, stream>>>(x0p, n4);
    }

    // 1. embeddings -> x0p rows 0,1
    embed_kernel<<<(DIM + 255) / 256, 256, 0, stream>>>(
        ray, emb1_w, emb1_b, emb2_w, emb2_b, x0p);

    // 2. qkv = x0 @ attn_in_w^T + b   (N = 3*DIM, plain [2, 3*DIM] out)
    gemv_wmma_kernel<DIM, false, false><<<(3 * DIM) / 16, 512, 0, stream>>>(
        x0p, attn_in_w, attn_in_b, nullptr, qkv, 3 * DIM, 3 * DIM);

    // 3. attention -> op rows 0,1
    attn_kernel<<<1, 256, 0, stream>>>(qkv, op);

    // 4. x1 = x0 + o @ attn_out_w^T + b
    gemv_wmma_kernel<DIM, false, true><<<DIM / 16, 512, 0, stream>>>(
        op, attn_out_w, attn_out_b, x0p, x1p, DIM, DIM);

    // 5. h = relu(x1 @ ffn1_w^T + b)
    gemv_wmma_kernel<DIM, true, false><<<DIM / 16, 512, 0, stream>>>(
        x1p, ffn1_w, ffn1_b, nullptr, hp, DIM, DIM);

    // 6. x2 = x1 + h @ ffn2_w^T + b
    gemv_wmma_kernel<DIM, false, true><<<DIM / 16, 512, 0, stream>>>(
        hp, ffn2_w, ffn2_b, x1p, x2p, DIM, DIM);

    // 7. a = relu(x2 @ exp_w1^T + b1)   (N = EDIM, plain [2, EDIM] out)
    gemv_wmma_kernel<DIM, true, false><<<EDIM / 16, 512, 0, stream>>>(
        x2p, exp_w1, exp_b1, nullptr, abuf, EDIM, EDIM);

    // 8. out = mean over seq of (a @ exp_w2^T + b2)
    final_kernel<<<1, 256, 0, stream>>>(abuf, exp_w2, exp_b2, out);
}